// P3P_RANSAC_40716289966080
// MI455X (gfx1250) — compile-verified
//
#include <hip/hip_runtime.h>

typedef __attribute__((ext_vector_type(2))) float v2f;
typedef __attribute__((ext_vector_type(8))) float v8f;

#define INL_TH_SQ 4.0f

// ---------------------------------------------------------------------------
// Kernel 1: per-model scores via V_WMMA_F32_16X16X4_F32.
// Each wave (wave32) owns 16 models; loops over all points in 16-wide tiles.
// A-tile  (16x4 f32): lane L -> M = L%16, VGPR0 = K=(L<16?0:2), VGPR1 = K+1
// B-tile  (4x16 f32): lane L -> N = L%16, VGPR0 = K=(L<16?0:2), VGPR1 = K+1
// D-tile (16x16 f32): lane L, VGPR r -> model r + 8*(L/16), point L%16
// ---------------------------------------------------------------------------
__global__ __launch_bounds__(128)
void p3p_score_kernel(const float* __restrict__ X,      // (N,4)
                      const float* __restrict__ xh,     // (N,3)
                      const float* __restrict__ models, // (B,3,4)
                      float* __restrict__ scores,       // (B,)
                      int N)
{
    const int lane  = threadIdx.x & 31;
    const int wave  = (blockIdx.x * blockDim.x + threadIdx.x) >> 5;
    const int m0    = wave << 4;                 // 16 models per wave
    const int mlane = m0 + (lane & 15);
    const int khalf = (lane >> 4) << 1;          // 0 (lanes 0-15) or 2 (16-31)

    // A-tiles: one per output coordinate row (u, v, w)
    const float* mp = models + mlane * 12;
    const v2f a0 = *(const v2f*)(mp + 0 + khalf);
    const v2f a1 = *(const v2f*)(mp + 4 + khalf);
    const v2f a2 = *(const v2f*)(mp + 8 + khalf);

    float acc[8];
#pragma unroll
    for (int r = 0; r < 8; ++r) acc[r] = 0.0f;

    for (int n0 = 0; n0 < N; n0 += 16) {
        const int nl = n0 + (lane & 15);
        const v2f b = *(const v2f*)(X + nl * 4 + khalf);   // B-tile (4x16)
        const float xx = xh[nl * 3 + 0];
        const float xy = xh[nl * 3 + 1];

        v8f zero = {};
        // D = A x B + 0 : projected u/v/w for 16 models x 16 points
        v8f U = __builtin_amdgcn_wmma_f32_16x16x4_f32(false, a0, false, b,
                                                      (short)0, zero, false, false);
        v8f V = __builtin_amdgcn_wmma_f32_16x16x4_f32(false, a1, false, b,
                                                      (short)0, zero, false, false);
        v8f W = __builtin_amdgcn_wmma_f32_16x16x4_f32(false, a2, false, b,
                                                      (short)0, zero, false, false);

#pragma unroll
        for (int r = 0; r < 8; ++r) {
            float w   = W[r];
            float rw  = __builtin_amdgcn_rcpf(w);
            float du  = __builtin_fmaf(U[r], rw, -xx);
            float dv  = __builtin_fmaf(V[r], rw, -xy);
            float err = __builtin_fmaf(du, du, dv * dv);
            err = fminf(err, INL_TH_SQ);           // clip(score, 0, th^2)
            err = (w > 0.0f) ? err : INL_TH_SQ;    // chirality mask
            acc[r] += err;
        }
    }

    // Reduce the 16 point-lanes inside each half-wave (xor masks stay in-half).
#pragma unroll
    for (int r = 0; r < 8; ++r) {
        float s = acc[r];
        s += __shfl_xor(s, 1, 32);
        s += __shfl_xor(s, 2, 32);
        s += __shfl_xor(s, 4, 32);
        s += __shfl_xor(s, 8, 32);
        if ((lane & 15) == 0)
            scores[m0 + r + ((lane >> 4) << 3)] = s;
    }
}

// ---------------------------------------------------------------------------
// Kernel 2: argmin over B model scores; emit best model + best score.
// ---------------------------------------------------------------------------
__global__ __launch_bounds__(256)
void p3p_argmin_kernel(const float* __restrict__ scores,
                       const float* __restrict__ models,
                       float* __restrict__ out,     // [0:12) model, [12+N] score
                       int* __restrict__ best_idx,
                       int B, int N)
{
    __shared__ float sv[256];
    __shared__ int   si[256];
    const int tid = threadIdx.x;

    float best = 3.4e38f;
    int   bi   = 0;
    for (int i = tid; i < B; i += 256) {
        float v = scores[i];
        if (v < best) { best = v; bi = i; }
    }
    sv[tid] = best; si[tid] = bi;
    __syncthreads();
    for (int s = 128; s > 0; s >>= 1) {
        if (tid < s && sv[tid + s] < sv[tid]) {
            sv[tid] = sv[tid + s];
            si[tid] = si[tid + s];
        }
        __syncthreads();
    }
    if (tid == 0) {
        int b = si[0];
        best_idx[0] = b;
        out[12 + N] = sv[0];
#pragma unroll
        for (int j = 0; j < 12; ++j) out[j] = models[b * 12 + j];
    }
}

// ---------------------------------------------------------------------------
// Kernel 3: inlier mask for the winning model (0.0 / 1.0 floats).
// ---------------------------------------------------------------------------
__global__ __launch_bounds__(256)
void p3p_inlier_kernel(const float* __restrict__ X,
                       const float* __restrict__ xh,
                       const float* __restrict__ models,
                       const int* __restrict__ best_idx,
                       float* __restrict__ out, int N)
{
    const int n = blockIdx.x * blockDim.x + threadIdx.x;
    if (n >= N) return;
    const float* M = models + best_idx[0] * 12;
    const float X0 = X[n * 4 + 0], X1 = X[n * 4 + 1];
    const float X2 = X[n * 4 + 2], X3 = X[n * 4 + 3];
    const float u = M[0] * X0 + M[1] * X1 + M[2]  * X2 + M[3]  * X3;
    const float v = M[4] * X0 + M[5] * X1 + M[6]  * X2 + M[7]  * X3;
    const float w = M[8] * X0 + M[9] * X1 + M[10] * X2 + M[11] * X3;
    const float du = u / w - xh[n * 3 + 0];
    const float dv = v / w - xh[n * 3 + 1];
    const float err = du * du + dv * dv;
    out[12 + n] = ((w > 0.0f) && (err < INL_TH_SQ)) ? 1.0f : 0.0f;
}

extern "C" void kernel_launch(void* const* d_in, const int* in_sizes, int n_in,
                              void* d_out, int out_size, void* d_ws, size_t ws_size,
                              hipStream_t stream) {
    const float* X      = (const float*)d_in[0];   // (N,4)
    const float* xh     = (const float*)d_in[1];   // (N,3)
    const float* models = (const float*)d_in[2];   // (B,3,4)
    float* out = (float*)d_out;

    const int N = in_sizes[0] / 4;                 // 8192
    const int B = in_sizes[2] / 12;                // 2048

    float* scores = (float*)d_ws;
    int* best_idx = (int*)((char*)d_ws + (size_t)B * sizeof(float));

    // 4 waves/block, 16 models/wave -> 64 models per block
    const int blocks1 = B / 64;
    p3p_score_kernel<<<blocks1, 128, 0, stream>>>(X, xh, models, scores, N);
    p3p_argmin_kernel<<<1, 256, 0, stream>>>(scores, models, out, best_idx, B, N);
    p3p_inlier_kernel<<<(N + 255) / 256, 256, 0, stream>>>(X, xh, models, best_idx, out, N);
}